// ResultParser_76141180223808
// MI455X (gfx1250) — compile-verified
//
#include <hip/hip_runtime.h>
#include <hip/hip_bf16.h>
#include <math.h>

#define NJ 22
#define CAM_DIM 3
#define PARAM_DIM (CAM_DIM + NJ*6 + 10)   // 145
#define KPAD 12                           // K=9 padded to 12 -> 3 x (K=4) WMMA steps
#define JQ 4                              // j-sub-tiles per wave (16 x 64 tile)

typedef __attribute__((ext_vector_type(2))) float v2f;
typedef __attribute__((ext_vector_type(8))) float v8f;

// ---------------------------------------------------------------------------
// Kernel 1: rot6d -> R (Gram-Schmidt), store K-padded rows + n2 transposed.
// Rpad layout: [N][NJ][KPAD] f32 (slots 9..11 zero).  n2T layout: [NJ][N] f32.
// ---------------------------------------------------------------------------
__global__ void prep_kernel(const float* __restrict__ params,
                            float* __restrict__ Rpad,
                            float* __restrict__ n2T, int n)
{
    int i = blockIdx.x * blockDim.x + threadIdx.x;
    if (i >= n) return;
    const float* p = params + (size_t)i * PARAM_DIM + CAM_DIM;
    for (int k = 0; k < NJ; ++k) {
        // x[n,k,r,c]: a1 = col0 (p[0],p[2],p[4]); a2 = col1 (p[1],p[3],p[5])
        float a1x = p[k*6+0], a1y = p[k*6+2], a1z = p[k*6+4];
        float a2x = p[k*6+1], a2y = p[k*6+3], a2z = p[k*6+5];
        float inv1 = rsqrtf(a1x*a1x + a1y*a1y + a1z*a1z);
        float b1x = a1x*inv1, b1y = a1y*inv1, b1z = a1z*inv1;
        float d   = b1x*a2x + b1y*a2y + b1z*a2z;
        float px  = a2x - d*b1x, py = a2y - d*b1y, pz = a2z - d*b1z;
        float inv2 = rsqrtf(px*px + py*py + pz*pz);
        float b2x = px*inv2, b2y = py*inv2, b2z = pz*inv2;
        float b3x = b1y*b2z - b1z*b2y;
        float b3y = b1z*b2x - b1x*b2z;
        float b3z = b1x*b2y - b1y*b2x;
        // columns are b1,b2,b3; row-major flatten (any consistent order works for G)
        float R[9] = { b1x,b2x,b3x,  b1y,b2y,b3y,  b1z,b2z,b3z };
        float* dst = Rpad + ((size_t)i*NJ + k) * KPAD;
        float n2 = 0.f;
        #pragma unroll
        for (int t = 0; t < 9; ++t) { dst[t] = R[t]; n2 += R[t]*R[t]; }
        dst[9] = 0.f; dst[10] = 0.f; dst[11] = 0.f;
        n2T[(size_t)k*n + i] = n2;
    }
}

// ---------------------------------------------------------------------------
// Kernel 2: one wave32 per 16x64 tile of score_map (4 j-sub-tiles, 4 accums).
// Per joint: A fragments + i-side n2 loaded ONCE, reused across the 4 j-sub-
// tiles; each sub-tile runs 3x v_wmma_f32_16x16x4_f32 (K=12 padded), then
// acc_q += sqrt(max(0, n2_i + n2_j - 2G)) via bare v_sqrt_f32 (1-ulp HW sqrt;
// result feeds a mean + threshold compare, full IEEE fixup is wasted VALU).
// Fused mask + score + store.
//
// A 16x4 f32 fragment (ISA 7.12.2): lanes 0-15 -> M=lane, v[0]=K0 v[1]=K1;
// lanes 16-31 -> M=lane-16, v[0]=K2 v[1]=K3.  B symmetric (N in lanes).
// C/D f32 16x16: lane 0-15 -> N=lane, VGPR r -> M=r; lanes 16-31 -> M=r+8.
// ---------------------------------------------------------------------------
__global__ __launch_bounds__(32) void tile_kernel(
    const float* __restrict__ Rpad, const float* __restrict__ n2T,
    const int* __restrict__ batch, const int* __restrict__ czyx,
    const float* __restrict__ top_score, float* __restrict__ score, int n)
{
    const int j0   = blockIdx.x * (16 * JQ);
    const int i0   = blockIdx.y * 16;
    const int lane = threadIdx.x;
    const int nIdx = lane & 15;
    const int hi   = lane >> 4;     // lane group
    const int kOfs = 2 * hi;        // fragment K offset for this lane group

    v8f acc[JQ] = {};               // per sub-tile: sum over joints of sqrt(d2)

    const float* RaBase = Rpad + (size_t)(i0 + nIdx) * NJ * KPAD + kOfs;
    const float* RbBase[JQ];
    #pragma unroll
    for (int q = 0; q < JQ; ++q)
        RbBase[q] = Rpad + (size_t)(j0 + q*16 + nIdx) * NJ * KPAD + kOfs;

    for (int k = 0; k < NJ; ++k) {
        // A fragments for this joint (reused by all 4 sub-tiles)
        const float* Ra = RaBase + (size_t)k * KPAD;
        v2f a0 = *(const v2f*)(Ra);
        v2f a1 = *(const v2f*)(Ra + 4);
        v2f a2 = *(const v2f*)(Ra + 8);

        // i-side n2 (reused by all 4 sub-tiles)
        const float4* pn = (const float4*)(n2T + (size_t)k*n + i0 + 8*hi);
        float4 nA = pn[0], nB = pn[1];
        float ni[8] = { nA.x,nA.y,nA.z,nA.w, nB.x,nB.y,nB.z,nB.w };

        #pragma unroll
        for (int q = 0; q < JQ; ++q) {
            const float* Rb = RbBase[q] + (size_t)k * KPAD;
            v2f b0 = *(const v2f*)(Rb);
            v2f b1 = *(const v2f*)(Rb + 4);
            v2f b2 = *(const v2f*)(Rb + 8);
            v8f c = {};
            c = __builtin_amdgcn_wmma_f32_16x16x4_f32(
                    false, a0, false, b0, (short)0, c, false, false);
            c = __builtin_amdgcn_wmma_f32_16x16x4_f32(
                    false, a1, false, b1, (short)0, c, false, false);
            c = __builtin_amdgcn_wmma_f32_16x16x4_f32(
                    false, a2, false, b2, (short)0, c, false, false);

            float n2j = n2T[(size_t)k*n + j0 + q*16 + nIdx];
            #pragma unroll
            for (int r = 0; r < 8; ++r) {
                float d2 = ni[r] + n2j - 2.0f * c[r];
                // bare v_sqrt_f32 (no IEEE 0.5-ulp refinement sequence)
                acc[q][r] += __builtin_amdgcn_sqrtf(fmaxf(d2, 0.0f));
            }
        }
    }

    const float inv = 1.0f / (float)NJ;

    // i-side (shared across sub-tiles) per-r data
    int   bi[8]; float ciy[8], cix[8];
    #pragma unroll
    for (int r = 0; r < 8; ++r) {
        int i = i0 + 8*hi + r;
        bi[r]  = batch[i];
        ciy[r] = (float)czyx[i*3+1];
        cix[r] = (float)czyx[i*3+2];
    }

    #pragma unroll
    for (int q = 0; q < JQ; ++q) {
        const int   j   = j0 + q*16 + nIdx;
        const int   bj  = batch[j];
        const float cyj = (float)czyx[j*3+1];
        const float cxj = (float)czyx[j*3+2];
        const float sj  = top_score[j];
        #pragma unroll
        for (int r = 0; r < 8; ++r) {
            int   i   = i0 + 8*hi + r;
            float dy  = ciy[r] - cyj;
            float dx  = cix[r] - cxj;
            float cd2 = dy*dy + dx*dx;           // ||c_i - c_j||^2 (exact, ints)
            float pd  = acc[q][r] * inv;
            bool  sim = (bi[r] == bj) && (cd2 <= 25.0f) && (pd < 2.5f);
            score[(size_t)i*n + j] = sim ? sj : 0.0f;
        }
    }
}

// ---------------------------------------------------------------------------
// Kernel 3: per-row first-argmax (jnp.argmax semantics) -> nms flag.
// ---------------------------------------------------------------------------
__global__ void nms_kernel(const float* __restrict__ score,
                           float* __restrict__ out, int n)
{
    __shared__ float sval[256];
    __shared__ int   sidx[256];
    const int row = blockIdx.x;
    const int tid = threadIdx.x;
    float best = -1.0f; int bestj = 0x7fffffff;
    for (int j = tid; j < n; j += blockDim.x) {
        float v = score[(size_t)row*n + j];
        if (v > best) { best = v; bestj = j; }   // strict > keeps first max
    }
    sval[tid] = best; sidx[tid] = bestj;
    __syncthreads();
    for (int s = 128; s > 0; s >>= 1) {
        if (tid < s) {
            float v = sval[tid+s]; int jj = sidx[tid+s];
            if (v > sval[tid] || (v == sval[tid] && jj < sidx[tid])) {
                sval[tid] = v; sidx[tid] = jj;
            }
        }
        __syncthreads();
    }
    if (tid == 0) out[row] = (sidx[0] == row) ? 1.0f : 0.0f;
}

extern "C" void kernel_launch(void* const* d_in, const int* in_sizes, int n_in,
                              void* d_out, int out_size, void* d_ws, size_t ws_size,
                              hipStream_t stream)
{
    (void)n_in; (void)out_size; (void)ws_size;
    const float* params = (const float*)d_in[0];
    const int*   batch  = (const int*)d_in[1];
    const int*   czyx   = (const int*)d_in[2];
    const float* tscore = (const float*)d_in[3];
    const int n = in_sizes[3];                 // N = 2048

    float* out_score = (float*)d_out;          // [n*n]
    float* out_nms   = out_score + (size_t)n*n;// [n]

    float* Rpad = (float*)d_ws;                      // n*NJ*KPAD f32
    float* n2T  = Rpad + (size_t)n * NJ * KPAD;      // NJ*n f32

    prep_kernel<<<(n + 127)/128, 128, 0, stream>>>(params, Rpad, n2T, n);

    dim3 grid(n/(16*JQ), n/16);
    tile_kernel<<<grid, 32, 0, stream>>>(Rpad, n2T, batch, czyx, tscore,
                                         out_score, n);

    nms_kernel<<<n, 256, 0, stream>>>(out_score, out_nms, n);
}